// Encoder_29712583754153
// MI455X (gfx1250) — compile-verified
//
#include <hip/hip_runtime.h>
#include <math.h>

#define BB 64
#define SS 80
#define KC 32
#define EE 256
#define HH 512
#define LL 128
#define H3 1536
#define HSTR 516   // padded LDS row stride (floats) to avoid bank conflicts

typedef float v2f __attribute__((ext_vector_type(2)));
typedef float v8f __attribute__((ext_vector_type(8)));

__device__ __forceinline__ v8f wmma4(v2f a, v2f b, v8f c) {
    // V_WMMA_F32_16X16X4_F32 : D = A(16x4,f32) * B(4x16,f32) + C(16x16,f32)
    return __builtin_amdgcn_wmma_f32_16x16x4_f32(false, a, false, b,
                                                 (short)0, c, false, false);
}

__device__ __forceinline__ float sigmoidf_(float x) {
    return 1.0f / (1.0f + __expf(-x));
}

// ---------------------------------------------------------------------------
// 1) stable argsort by descending length (B=64, one block)
// ---------------------------------------------------------------------------
__global__ void sort_kernel(const int* __restrict__ len,
                            int* __restrict__ sidx, int* __restrict__ slen) {
    int i = threadIdx.x;            // 0..63
    int li = len[i];
    int rank = 0;
    for (int j = 0; j < BB; ++j) {
        int lj = len[j];
        if (lj > li || (lj == li && j < i)) rank++;
    }
    sidx[rank] = i;
    slen[rank] = li;
}

// ---------------------------------------------------------------------------
// 2) multi-hot embedding with dedupe + tanh.  One block per (sorted b, s).
//    emb laid out [S][B][E] so GEMM M-rows (m = s*64+b) are contiguous.
// ---------------------------------------------------------------------------
__global__ void embed_kernel(const int* __restrict__ seq,
                             const int* __restrict__ sidx,
                             const float* __restrict__ EW,
                             float* __restrict__ emb) {
    __shared__ int codes[KC];
    __shared__ int valid[KC];
    int b = blockIdx.x / SS;        // sorted position
    int s = blockIdx.x % SS;
    int tid = threadIdx.x;          // 0..255
    int srow = sidx[b];             // original patient index
    if (tid < KC) codes[tid] = seq[((size_t)srow * SS + s) * KC + tid];
    __syncthreads();
    if (tid < KC) {
        int c = codes[tid];
        int v = (c != 0);
        for (int j = 0; j < tid; ++j)
            if (codes[j] == c) v = 0;   // duplicates collapse to one
        valid[tid] = v;
    }
    __syncthreads();
    float acc = 0.0f;
    for (int k = 0; k < KC; ++k)
        if (valid[k]) acc += EW[(size_t)codes[k] * EE + tid];
    emb[((size_t)s * BB + b) * EE + tid] = tanhf(acc);
}

// ---------------------------------------------------------------------------
// 3) transpose W[H3][C] -> WT[C][H3] so WMMA B-operand loads are coalesced
// ---------------------------------------------------------------------------
__global__ void transpose_k(const float* __restrict__ W,
                            float* __restrict__ WT, int C) {
    int idx = blockIdx.x * blockDim.x + threadIdx.x;
    if (idx >= H3 * C) return;
    int n = idx / C;
    int c = idx - n * C;
    WT[(size_t)c * H3 + n] = W[idx];
}

// ---------------------------------------------------------------------------
// 4) GI[s][b][3H] = emb @ Wi^T + bi  (M=5120, N=1536, K=256) — fp32 WMMA
//    one 16x16 tile per wave, K stepped by 4
// ---------------------------------------------------------------------------
__global__ void gi_gemm(const float* __restrict__ emb,
                        const float* __restrict__ WiT,
                        const float* __restrict__ bi,
                        float* __restrict__ GI) {
    int wid  = (blockIdx.x * blockDim.x + threadIdx.x) >> 5;
    int lane = threadIdx.x & 31;
    int half = lane >> 4, ln = lane & 15;
    int mt = wid / 96, nt = wid - mt * 96;
    int m0 = mt * 16, n0 = nt * 16;

    v8f acc = {0,0,0,0,0,0,0,0};
    const float* arow = emb + (size_t)(m0 + ln) * EE;
    for (int k0 = 0; k0 < EE; k0 += 4) {
        int k = k0 + 2 * half;
        v2f a; a.x = arow[k]; a.y = arow[k + 1];
        const float* bp = WiT + (size_t)k * H3 + n0 + ln;
        v2f b; b.x = bp[0]; b.y = bp[H3];
        acc = wmma4(a, b, acc);
    }
    float bn = bi[n0 + ln];
    for (int r = 0; r < 8; ++r) {
        int m = m0 + r + 8 * half;
        GI[(size_t)m * H3 + n0 + ln] = acc[r] + bn;
    }
}

// ---------------------------------------------------------------------------
// 5) full 80-step GRU scan + latent projection.
//    4 workgroups, each owns 16 batch rows; h lives in LDS; no inter-WG sync.
//    Wave w owns N-tiles {w, w+16, w+32, w+48, w+64, w+80} so each lane holds
//    matching (r,z,n) gate triples in registers.
// ---------------------------------------------------------------------------
__global__ void __launch_bounds__(512)
gru_scan(const float* __restrict__ GI, const float* __restrict__ WhT,
         const float* __restrict__ bh, const int* __restrict__ slen,
         const float* __restrict__ Wlat, const float* __restrict__ blat,
         float* __restrict__ out) {
    __shared__ float hs[16 * HSTR];
    __shared__ int lens[16];
    int tid  = threadIdx.x;
    int r0   = blockIdx.x * 16;
    int w    = tid >> 5;
    int lane = tid & 31, half = lane >> 4, ln = lane & 15;

    for (int i = tid; i < 16 * HSTR; i += 512) hs[i] = 0.0f;
    if (tid < 16) lens[tid] = slen[r0 + tid];
    __syncthreads();

    const int j1 = 16 * w + ln;      // gate column, group 1 (tile w)
    const int j2 = j1 + 256;         // gate column, group 2 (tile w+16)

    for (int t = 0; t < SS; ++t) {
        v8f c0 = {0,0,0,0,0,0,0,0}, c1 = {0,0,0,0,0,0,0,0};
        v8f c2 = {0,0,0,0,0,0,0,0}, c3 = {0,0,0,0,0,0,0,0};
        v8f c4 = {0,0,0,0,0,0,0,0}, c5 = {0,0,0,0,0,0,0,0};
        // gh = h @ Wh^T  for this WG's 16 rows, all 1536 cols split over waves
        for (int k0 = 0; k0 < HH; k0 += 4) {
            int k = k0 + 2 * half;
            v2f a; a.x = hs[ln * HSTR + k]; a.y = hs[ln * HSTR + k + 1];
            const float* bp = WhT + (size_t)k * H3 + j1;   // + n0 offsets below
            v2f b0; b0.x = bp[0];    b0.y = bp[H3];
            v2f b1; b1.x = bp[256];  b1.y = bp[H3 + 256];
            v2f b2; b2.x = bp[512];  b2.y = bp[H3 + 512];
            v2f b3; b3.x = bp[768];  b3.y = bp[H3 + 768];
            v2f b4; b4.x = bp[1024]; b4.y = bp[H3 + 1024];
            v2f b5; b5.x = bp[1280]; b5.y = bp[H3 + 1280];
            c0 = wmma4(a, b0, c0);
            c1 = wmma4(a, b1, c1);
            c2 = wmma4(a, b2, c2);
            c3 = wmma4(a, b3, c3);
            c4 = wmma4(a, b4, c4);
            c5 = wmma4(a, b5, c5);
        }
        __syncthreads();   // all reads of hs done before gate-phase writes

        float bhr1 = bh[j1], bhz1 = bh[j1 + 512], bhn1 = bh[j1 + 1024];
        float bhr2 = bh[j2], bhz2 = bh[j2 + 512], bhn2 = bh[j2 + 1024];
        for (int r = 0; r < 8; ++r) {
            int m = r + 8 * half;
            const float* gip = GI + ((size_t)t * BB + r0 + m) * H3;
            bool upd = t < lens[m];
            {   // columns j1: r from c0, z from c2, n from c4
                float rr = sigmoidf_(gip[j1]        + c0[r] + bhr1);
                float zz = sigmoidf_(gip[j1 + 512]  + c2[r] + bhz1);
                float nn = tanhf   (gip[j1 + 1024] + rr * (c4[r] + bhn1));
                float hp = hs[m * HSTR + j1];
                if (upd) hs[m * HSTR + j1] = (1.0f - zz) * nn + zz * hp;
            }
            {   // columns j2: r from c1, z from c3, n from c5
                float rr = sigmoidf_(gip[j2]        + c1[r] + bhr2);
                float zz = sigmoidf_(gip[j2 + 512]  + c3[r] + bhz2);
                float nn = tanhf   (gip[j2 + 1024] + rr * (c5[r] + bhn2));
                float hp = hs[m * HSTR + j2];
                if (upd) hs[m * HSTR + j2] = (1.0f - zz) * nn + zz * hp;
            }
        }
        __syncthreads();   // gate writes visible before next step's GEMM
    }

    // latent: out = tanh(hT @ Wlat^T + blat)
    for (int idx = tid; idx < 16 * LL; idx += 512) {
        int m = idx >> 7, l = idx & (LL - 1);
        const float* wl = Wlat + (size_t)l * HH;
        float acc = blat[l];
        for (int j = 0; j < HH; ++j) acc += hs[m * HSTR + j] * wl[j];
        out[(size_t)(r0 + m) * LL + l] = tanhf(acc);
    }
}

// ---------------------------------------------------------------------------
extern "C" void kernel_launch(void* const* d_in, const int* in_sizes, int n_in,
                              void* d_out, int out_size, void* d_ws, size_t ws_size,
                              hipStream_t stream) {
    const int*   seq  = (const int*)d_in[0];
    const int*   len  = (const int*)d_in[1];
    const float* EW   = (const float*)d_in[2];
    const float* Wi   = (const float*)d_in[3];
    const float* Wh   = (const float*)d_in[4];
    const float* bi   = (const float*)d_in[5];
    const float* bh   = (const float*)d_in[6];
    const float* Wlat = (const float*)d_in[7];
    const float* blat = (const float*)d_in[8];
    float* out = (float*)d_out;

    char* ws = (char*)d_ws;
    int*   sidx = (int*)ws;                      // 64 ints
    int*   slen = (int*)(ws + 256);              // 64 ints
    size_t off  = 512;
    float* emb  = (float*)(ws + off); off += sizeof(float) * (size_t)SS * BB * EE;
    float* WiT  = (float*)(ws + off); off += sizeof(float) * (size_t)EE * H3;
    float* WhT  = (float*)(ws + off); off += sizeof(float) * (size_t)HH * H3;
    float* GI   = (float*)(ws + off);            // S*B*3H floats

    sort_kernel<<<1, 64, 0, stream>>>(len, sidx, slen);
    embed_kernel<<<BB * SS, 256, 0, stream>>>(seq, sidx, EW, emb);
    transpose_k<<<(H3 * EE + 255) / 256, 256, 0, stream>>>(Wi, WiT, EE);
    transpose_k<<<(H3 * HH + 255) / 256, 256, 0, stream>>>(Wh, WhT, HH);
    gi_gemm<<<(320 * 96) / 8, 256, 0, stream>>>(emb, WiT, bi, GI);
    gru_scan<<<4, 512, 0, stream>>>(GI, WhT, bh, slen, Wlat, blat, out);
}